// Decoder_10668698763754
// MI455X (gfx1250) — compile-verified
//
#include <hip/hip_runtime.h>
#include <math.h>

typedef __attribute__((ext_vector_type(2))) float v2f;
typedef __attribute__((ext_vector_type(8))) float v8f;

// Problem constants (fixed by the reference)
#define BB 32
#define NN 200000
#define XSZ 256
#define KX 129            // XS/2 + 1
#define SCALE_C 100.0f
#define CENTER_C 128.0f
#define W0_FIRST_C 30.0f

#define K1_CHUNKS 32
#define K1_THREADS 256
#define NPC (NN / K1_CHUNKS)   // 6250

#define TWO_PI_OVER_256 0.0245436926061702596f  // 2*pi/256
#define ARS 260                 // padded LDS row stride (260 % 64 != 0 -> no bank aliasing)
#define DFT_ROWS (BB * KX)      // 4128 = 32*129 spectral rows for the column transforms

__device__ __forceinline__ void euler_zyz_dev(const float* ang, float* R) {
    float a = ang[0], b = ang[1], g = ang[2];
    float ca = cosf(a), sa = sinf(a);
    float cb = cosf(b), sb = sinf(b);
    float cg = cosf(g), sg = sinf(g);
    R[0] =  ca * cb * cg - sa * sg;
    R[1] = -ca * cb * sg - sa * cg;
    R[2] =  ca * sb;
    R[3] =  sa * cb * cg + ca * sg;
    R[4] = -sa * cb * sg + ca * cg;
    R[5] =  sa * sb;
    R[6] = -sb * cg;
    R[7] =  sb * sg;
    R[8] =  cb;
}

// -------- K1: partial reduction for h_pre[b,m] = sum_n sum_i crot[b,n,i]*W1[(3n+i)*8+m]
__global__ void k1_partial(const float* __restrict__ rows,
                           const float* __restrict__ coords,
                           const float* __restrict__ W1,
                           float* __restrict__ partials) {
    int b = blockIdx.x >> 5;        // 32 chunks per batch
    int chunk = blockIdx.x & 31;
    float R[9];
    euler_zyz_dev(rows + b * 3, R);
    float acc[8];
#pragma unroll
    for (int m = 0; m < 8; ++m) acc[m] = 0.0f;

    int n0 = chunk * NPC;
    for (int n = n0 + threadIdx.x; n < n0 + NPC; n += K1_THREADS) {
        float c0 = coords[n * 3 + 0];
        float c1 = coords[n * 3 + 1];
        float c2 = coords[n * 3 + 2];
        float x = R[0] * c0 + R[1] * c1 + R[2] * c2;
        float y = R[3] * c0 + R[4] * c1 + R[5] * c2;
        float z = R[6] * c0 + R[7] * c1 + R[8] * c2;
        const float* w = W1 + (size_t)n * 24;   // rows 3n, 3n+1, 3n+2 of (3N,8)
        __builtin_prefetch(w + (size_t)24 * K1_THREADS, 0, 1);  // global_prefetch_b8
#pragma unroll
        for (int m = 0; m < 8; ++m)
            acc[m] += x * w[m] + y * w[8 + m] + z * w[16 + m];
    }

    __shared__ float red[K1_THREADS];
    for (int m = 0; m < 8; ++m) {
        red[threadIdx.x] = acc[m];
        __syncthreads();
        for (int s = K1_THREADS / 2; s > 0; s >>= 1) {
            if (threadIdx.x < s) red[threadIdx.x] += red[threadIdx.x + s];
            __syncthreads();
        }
        if (threadIdx.x == 0)
            partials[((size_t)b * K1_CHUNKS + chunk) * 8 + m] = red[0];
        __syncthreads();
    }
}

// -------- K2: finalize h = sin(30*(sum + b1)); then 3 hidden sin layers (8x8). One block, 256 threads.
__global__ void k2_finish(const float* __restrict__ partials,
                          const float* __restrict__ b1,
                          const float* __restrict__ Wh,
                          const float* __restrict__ bh,
                          float* __restrict__ hout) {
    __shared__ float hb[BB * 8];
    __shared__ float hn[BB * 8];
    int t = threadIdx.x;            // 0..255
    int b = t >> 3, m = t & 7;
    float s = 0.0f;
    for (int c = 0; c < K1_CHUNKS; ++c)
        s += partials[((size_t)b * K1_CHUNKS + c) * 8 + m];
    hb[t] = sinf(W0_FIRST_C * (s + b1[m]));
    __syncthreads();
    for (int L = 0; L < 3; ++L) {
        float a = bh[L * 8 + m];
#pragma unroll
        for (int k = 0; k < 8; ++k)
            a += hb[b * 8 + k] * Wh[L * 64 + k * 8 + m];
        hn[t] = sinf(a);
        __syncthreads();
        hb[t] = hn[t];
        __syncthreads();
    }
    hout[t] = hb[t];
}

// -------- K3: delta[b,n] = h[b,:]·W5[:,n] + b5[n] via V_WMMA_F32_16X16X4_F32
// M = 16 batches (2 tiles for B=32), N-tile = 16 points, K = 8 (2 chunks of 4).
__global__ void k3_delta_wmma(const float* __restrict__ h,
                              const float* __restrict__ W5,
                              const float* __restrict__ b5,
                              float* __restrict__ delta) {
    __shared__ float hs[BB * 8];
    int t = threadIdx.x;                 // 128 threads = 4 waves
    for (int i = t; i < BB * 8; i += 128) hs[i] = h[i];
    __syncthreads();

    int wave = t >> 5;
    int lane = t & 31;
    int ntile = blockIdx.x * 4 + wave;   // 3125 blocks * 4 = 12500 tiles (N/16)
    int n0 = ntile * 16;
    int col = lane & 15;                 // A-row (batch) for A, column n for B/D
    int hi = lane >> 4;                  // 0 -> K 0,1 ; 1 -> K 2,3 (within 4-chunk)

    v8f acc0 = {};
    v8f acc1 = {};
#pragma unroll
    for (int kc = 0; kc < 2; ++kc) {
        int kbase = kc * 4 + hi * 2;
        v2f bvec;
        bvec.x = W5[(size_t)kbase * NN + n0 + col];
        bvec.y = W5[(size_t)(kbase + 1) * NN + n0 + col];
        v2f a0, a1;
        a0.x = hs[col * 8 + kbase];
        a0.y = hs[col * 8 + kbase + 1];
        a1.x = hs[(16 + col) * 8 + kbase];
        a1.y = hs[(16 + col) * 8 + kbase + 1];
        acc0 = __builtin_amdgcn_wmma_f32_16x16x4_f32(false, a0, false, bvec,
                                                     (short)0, acc0, false, false);
        acc1 = __builtin_amdgcn_wmma_f32_16x16x4_f32(false, a1, false, bvec,
                                                     (short)0, acc1, false, false);
    }
    float bb = b5[n0 + col];
#pragma unroll
    for (int r = 0; r < 8; ++r) {
        int brow = r + hi * 8;   // D row = batch index within tile
        delta[(size_t)brow * NN + n0 + col] = acc0[r] + bb;
        delta[(size_t)(16 + brow) * NN + n0 + col] = acc1[r] + bb;
    }
}

// -------- K4: per-image bilinear scatter into 256KB LDS image + fused 7x7 Gaussian blur.
__global__ void __launch_bounds__(1024)
k4_scatter_blur(const float* __restrict__ rows,
                const float* __restrict__ shifts,
                const float* __restrict__ coords,
                const float* __restrict__ delta,
                float* __restrict__ img) {
    extern __shared__ float simg[];     // XSZ*XSZ floats = 256 KB (CDNA5: 320KB LDS/WGP)
    int b = blockIdx.x;
    int t = threadIdx.x;
    for (int i = t; i < XSZ * XSZ; i += 1024) simg[i] = 0.0f;
    float R[9];
    euler_zyz_dev(rows + b * 3, R);
    float sx = shifts[b * 2 + 0];
    float sy = shifts[b * 2 + 1];
    __syncthreads();

    for (int n = t; n < NN; n += 1024) {
        float c0 = coords[n * 3 + 0];
        float c1 = coords[n * 3 + 1];
        float c2 = coords[n * 3 + 2];
        float cx = R[0] * c0 + R[1] * c1 + R[2] * c2;
        float cy = R[3] * c0 + R[4] * c1 + R[5] * c2;
        float d = delta[(size_t)b * NN + n];
        float x = cx * SCALE_C + CENTER_C - sx;
        float y = cy * SCALE_C + CENTER_C - sy;
        float x0f = floorf(x), y0f = floorf(y);
        float fx = x - x0f, fy = y - y0f;
        int x0 = (int)x0f, y0 = (int)y0f;
        float w00 = (1.0f - fx) * (1.0f - fy) * d;
        float w10 = fx * (1.0f - fy) * d;
        float w01 = (1.0f - fx) * fy * d;
        float w11 = fx * fy * d;
        if (x0 >= 0 && x0 < XSZ && y0 >= 0 && y0 < XSZ)       atomicAdd(&simg[y0 * XSZ + x0], w00);
        if (x0 + 1 >= 0 && x0 + 1 < XSZ && y0 >= 0 && y0 < XSZ) atomicAdd(&simg[y0 * XSZ + x0 + 1], w10);
        if (x0 >= 0 && x0 < XSZ && y0 + 1 >= 0 && y0 + 1 < XSZ) atomicAdd(&simg[(y0 + 1) * XSZ + x0], w01);
        if (x0 + 1 >= 0 && x0 + 1 < XSZ && y0 + 1 >= 0 && y0 + 1 < XSZ) atomicAdd(&simg[(y0 + 1) * XSZ + x0 + 1], w11);
    }
    __syncthreads();

    // 7x7 separable Gaussian with zero padding (== vertical then horizontal SAME conv)
    float kk[7];
    float ksum = 0.0f;
#pragma unroll
    for (int i = 0; i < 7; ++i) { float u = (float)(i - 3); kk[i] = expf(-0.5f * u * u); ksum += kk[i]; }
#pragma unroll
    for (int i = 0; i < 7; ++i) kk[i] /= ksum;

    for (int p = t; p < XSZ * XSZ; p += 1024) {
        int y = p >> 8, x = p & 255;
        float acc = 0.0f;
#pragma unroll
        for (int dy = -3; dy <= 3; ++dy) {
            int yy = y + dy;
            if (yy < 0 || yy >= XSZ) continue;
            float wy = kk[dy + 3];
#pragma unroll
            for (int dx = -3; dx <= 3; ++dx) {
                int xx = x + dx;
                if (xx < 0 || xx >= XSZ) continue;
                acc += wy * kk[dx + 3] * simg[yy * XSZ + xx];
            }
        }
        img[(size_t)b * XSZ * XSZ + p] = acc;
    }
}

// -------- K5: rfft along x. Fr layout [b][y][k][2], k = 0..128
__global__ void k5_rowfft(const float* __restrict__ img, float* __restrict__ Fr) {
    __shared__ float row[XSZ];
    __shared__ float ct[XSZ];
    __shared__ float st[XSZ];
    int t = threadIdx.x;            // 256
    int b = blockIdx.x >> 8;
    int y = blockIdx.x & 255;
    row[t] = img[((size_t)b * XSZ + y) * XSZ + t];
    float ang = TWO_PI_OVER_256 * (float)t;
    ct[t] = cosf(ang);
    st[t] = sinf(ang);
    __syncthreads();
    if (t <= 128) {
        float re = 0.0f, im = 0.0f;
        for (int x = 0; x < XSZ; ++x) {
            int idx = (t * x) & 255;
            float v = row[x];
            re += v * ct[idx];
            im -= v * st[idx];
        }
        size_t o = (((size_t)b * XSZ + y) * KX + t) * 2;
        Fr[o] = re;
        Fr[o + 1] = im;
    }
}

// -------- K6 (WMMA): full complex FFT along y (e^{-i t}) + ctf multiply.
// GEMM view: 4128 (b,k)-rows x K=256 (y) x N=256 (ky).
// Complex via 4 real accumulator chains:  Re = Ar@C + Ai@S,  Im = Ai@C - Ar@S.
// Output G layout [b][k][ky][2].
__global__ void k6_colfft_wmma(const float* __restrict__ Fr,
                               const float* __restrict__ ctf,
                               float* __restrict__ G) {
    __shared__ float Ar[16 * ARS];
    __shared__ float Ai[16 * ARS];
    __shared__ float ct[XSZ];
    __shared__ float st[XSZ];
    int t = threadIdx.x;            // 256 = 8 waves
    int rowbase = blockIdx.x * 16;  // 258 blocks

    float ang = TWO_PI_OVER_256 * (float)t;
    ct[t] = cosf(ang);
    st[t] = sinf(ang);
    for (int idx = t; idx < 16 * XSZ; idx += 256) {
        int r = idx >> 8;
        int y = idx & 255;
        int row = rowbase + r;
        int b = row / KX;
        int k = row % KX;
        size_t o = (((size_t)b * XSZ + y) * KX + k) * 2;
        Ar[r * ARS + y] = Fr[o];
        Ai[r * ARS + y] = Fr[o + 1];
    }
    __syncthreads();

    int wave = t >> 5;
    int lane = t & 31;
    int col = lane & 15;            // A row (M) and B/D column (N)
    int hi = lane >> 4;             // K sub-pair select within each 4-chunk

    for (int tt = 0; tt < 2; ++tt) {
        int kybase = (wave + tt * 8) * 16;   // 16 ky-tiles over 8 waves x 2
        int kyAbs = kybase + col;
        v8f P = {}, Q = {}, Rp = {}, Sp = {};
#pragma unroll 4
        for (int c = 0; c < 64; ++c) {
            int yb = 4 * c + 2 * hi;
            v2f ar, ai, bc, bs;
            ar.x = Ar[col * ARS + yb];
            ar.y = Ar[col * ARS + yb + 1];
            ai.x = Ai[col * ARS + yb];
            ai.y = Ai[col * ARS + yb + 1];
            int i0 = (yb * kyAbs) & 255;
            int i1 = ((yb + 1) * kyAbs) & 255;
            bc.x = ct[i0]; bc.y = ct[i1];
            bs.x = st[i0]; bs.y = st[i1];
            P  = __builtin_amdgcn_wmma_f32_16x16x4_f32(false, ar, false, bc, (short)0, P,  false, false);
            Q  = __builtin_amdgcn_wmma_f32_16x16x4_f32(false, ai, false, bs, (short)0, Q,  false, false);
            Rp = __builtin_amdgcn_wmma_f32_16x16x4_f32(false, ai, false, bc, (short)0, Rp, false, false);
            Sp = __builtin_amdgcn_wmma_f32_16x16x4_f32(false, ar, false, bs, (short)0, Sp, false, false);
        }
#pragma unroll
        for (int r = 0; r < 8; ++r) {
            int row = rowbase + r + 8 * hi;
            int b = row / KX;
            int k = row % KX;
            float f = ctf[((size_t)b * XSZ + kyAbs) * KX + k];
            float re = (P[r] + Q[r]) * f;
            float im = (Rp[r] - Sp[r]) * f;
            size_t go = (((size_t)b * KX + k) * XSZ + kyAbs) * 2;
            G[go] = re;
            G[go + 1] = im;
        }
    }
}

// -------- K7 (WMMA): inverse complex FFT along y (e^{+i t}, 1/256 scale).
// Re = Gr@C - Gi@S,  Im = Gr@S + Gi@C.  Input rows of G are contiguous. Output H [b][y][k][2].
__global__ void k7_colifft_wmma(const float* __restrict__ G, float* __restrict__ H) {
    __shared__ float Ar[16 * ARS];
    __shared__ float Ai[16 * ARS];
    __shared__ float ct[XSZ];
    __shared__ float st[XSZ];
    int t = threadIdx.x;            // 256 = 8 waves
    int rowbase = blockIdx.x * 16;  // 258 blocks

    float ang = TWO_PI_OVER_256 * (float)t;
    ct[t] = cosf(ang);
    st[t] = sinf(ang);
    for (int idx = t; idx < 16 * XSZ; idx += 256) {
        int r = idx >> 8;
        int y = idx & 255;
        int row = rowbase + r;
        size_t o = ((size_t)row * XSZ + y) * 2;   // G rows are contiguous: row = b*KX + k
        Ar[r * ARS + y] = G[o];
        Ai[r * ARS + y] = G[o + 1];
    }
    __syncthreads();

    int wave = t >> 5;
    int lane = t & 31;
    int col = lane & 15;
    int hi = lane >> 4;

    for (int tt = 0; tt < 2; ++tt) {
        int ybase = (wave + tt * 8) * 16;
        int yAbs = ybase + col;
        v8f P = {}, Q = {}, Rp = {}, Sp = {};
#pragma unroll 4
        for (int c = 0; c < 64; ++c) {
            int kb = 4 * c + 2 * hi;
            v2f ar, ai, bc, bs;
            ar.x = Ar[col * ARS + kb];
            ar.y = Ar[col * ARS + kb + 1];
            ai.x = Ai[col * ARS + kb];
            ai.y = Ai[col * ARS + kb + 1];
            int i0 = (kb * yAbs) & 255;
            int i1 = ((kb + 1) * yAbs) & 255;
            bc.x = ct[i0]; bc.y = ct[i1];
            bs.x = st[i0]; bs.y = st[i1];
            P  = __builtin_amdgcn_wmma_f32_16x16x4_f32(false, ar, false, bc, (short)0, P,  false, false);
            Q  = __builtin_amdgcn_wmma_f32_16x16x4_f32(false, ai, false, bs, (short)0, Q,  false, false);
            Rp = __builtin_amdgcn_wmma_f32_16x16x4_f32(false, ar, false, bs, (short)0, Rp, false, false);
            Sp = __builtin_amdgcn_wmma_f32_16x16x4_f32(false, ai, false, bc, (short)0, Sp, false, false);
        }
#pragma unroll
        for (int r = 0; r < 8; ++r) {
            int row = rowbase + r + 8 * hi;
            int b = row / KX;
            int k = row % KX;
            float re = (P[r] - Q[r]) * (1.0f / 256.0f);
            float im = (Rp[r] + Sp[r]) * (1.0f / 256.0f);
            size_t ho = (((size_t)b * XSZ + yAbs) * KX + k) * 2;
            H[ho] = re;
            H[ho + 1] = im;
        }
    }
}

// -------- K8: irfft along x (Hermitian half-spectrum, with 1/256) -> output
__global__ void k8_rowirfft(const float* __restrict__ H, float* __restrict__ out) {
    __shared__ float hr[KX];
    __shared__ float hi[KX];
    __shared__ float ct[XSZ];
    __shared__ float st[XSZ];
    int t = threadIdx.x;            // 256 -> x
    int b = blockIdx.x >> 8;
    int y = blockIdx.x & 255;
    if (t < KX) {
        size_t o = (((size_t)b * XSZ + y) * KX + t) * 2;
        hr[t] = H[o];
        hi[t] = H[o + 1];
    }
    float ang = TWO_PI_OVER_256 * (float)t;
    ct[t] = cosf(ang);
    st[t] = sinf(ang);
    __syncthreads();
    int x = t;
    float acc = hr[0] + (((x & 1) != 0) ? -hr[128] : hr[128]);
    for (int k = 1; k < 128; ++k) {
        int idx = (k * x) & 255;
        acc += 2.0f * (hr[k] * ct[idx] - hi[k] * st[idx]);
    }
    out[((size_t)b * XSZ + y) * XSZ + x] = acc * (1.0f / 256.0f);
}

extern "C" void kernel_launch(void* const* d_in, const int* in_sizes, int n_in,
                              void* d_out, int out_size, void* d_ws, size_t ws_size,
                              hipStream_t stream) {
    const float* rows   = (const float*)d_in[0];
    const float* shifts = (const float*)d_in[1];
    const float* coords = (const float*)d_in[2];
    const float* W1     = (const float*)d_in[3];
    const float* b1     = (const float*)d_in[4];
    const float* Wh     = (const float*)d_in[5];
    const float* bh     = (const float*)d_in[6];
    const float* W5     = (const float*)d_in[7];
    const float* b5     = (const float*)d_in[8];
    const float* ctf    = (const float*)d_in[9];
    float* outp = (float*)d_out;

    float* ws = (float*)d_ws;
    float* partials = ws;                               // 32*32*8      = 8192
    float* hbuf     = partials + BB * K1_CHUNKS * 8;    // 256
    float* delta    = hbuf + BB * 8;                    // 32*200000    = 6.4M
    float* img      = delta + (size_t)BB * NN;          // 32*256*256   = 2.10M
    float* Fr       = img + (size_t)BB * XSZ * XSZ;     // 32*256*129*2 = 2.11M
    float* G        = Fr + (size_t)BB * XSZ * KX * 2;   // 2.11M
    float* H        = G + (size_t)BB * KX * XSZ * 2;    // 2.11M
    (void)in_sizes; (void)n_in; (void)out_size; (void)ws_size;

    // allow 256KB dynamic LDS for the scatter+blur kernel
    hipFuncSetAttribute(reinterpret_cast<const void*>(k4_scatter_blur),
                        hipFuncAttributeMaxDynamicSharedMemorySize,
                        XSZ * XSZ * (int)sizeof(float));

    k1_partial<<<BB * K1_CHUNKS, K1_THREADS, 0, stream>>>(rows, coords, W1, partials);
    k2_finish<<<1, 256, 0, stream>>>(partials, b1, Wh, bh, hbuf);
    k3_delta_wmma<<<(NN / 16) / 4, 128, 0, stream>>>(hbuf, W5, b5, delta);
    k4_scatter_blur<<<BB, 1024, XSZ * XSZ * sizeof(float), stream>>>(rows, shifts, coords, delta, img);
    k5_rowfft<<<BB * XSZ, 256, 0, stream>>>(img, Fr);
    k6_colfft_wmma<<<DFT_ROWS / 16, 256, 0, stream>>>(Fr, ctf, G);
    k7_colifft_wmma<<<DFT_ROWS / 16, 256, 0, stream>>>(G, H);
    k8_rowirfft<<<BB * XSZ, 256, 0, stream>>>(H, outp);
}